// MyModel_61933428411467
// MI455X (gfx1250) — compile-verified
//
#include <hip/hip_runtime.h>

// One wave32. 15 floats in, 5 floats out: pure launch-latency-bound problem.
// V_WMMA_F32_16X16X4_F32 is used as the cross-lane reduction engine for the
// two broadcast means (A = per-sample node values, B = all-ones (layout-
// invariant), C = 5th sample). Everything else is VALU f32, matching the
// reference's logsumexp numerics.

typedef float v2f __attribute__((ext_vector_type(2)));
typedef float v8f __attribute__((ext_vector_type(8)));

__device__ __forceinline__ float lse2(float a, float b) {
    float m = fmaxf(a, b);
    return m + __logf(__expf(a - m) + __expf(b - m));
}

__global__ void __launch_bounds__(32)
circuit_wmma_kernel(const float* __restrict__ lp, float* __restrict__ out) {
    const int l    = threadIdx.x;  // 0..31
    const int m    = l & 15;       // A-matrix row M (lane within half)
    const int half = l >> 4;       // 0 -> K={0,1}, 1 -> K={2,3}
    const int k0   = 2 * half;     // sample index for a.x
    const int k1   = 2 * half + 1; // sample index for a.y

    // node5(k) = LSE(s,s) = s + ln2 ; node7(k) = LSE(s, -2000), s = lp1+lp2
    auto n5 = [&](int k) { float s = lp[k * 3 + 1] + lp[k * 3 + 2]; return lse2(s, s); };
    auto n7 = [&](int k) { float s = lp[k * 3 + 1] + lp[k * 3 + 2]; return lse2(s, -2000.0f); };

    // Every lane computes the values it must hold for the documented A layout.
    float n5_k0 = n5(k0), n7_k0 = n7(k0);
    float n5_k1 = n5(k1), n7_k1 = n7(k1);
    float n5_4  = n5(4),  n7_4  = n7(4);   // 5th sample goes into C

    // A (16x4 f32): row 0 = node5 samples 0..3, row 1 = node7 samples 0..3.
    v2f a;
    a.x = (m == 0) ? n5_k0 : ((m == 1) ? n7_k0 : 0.0f);
    a.y = (m == 0) ? n5_k1 : ((m == 1) ? n7_k1 : 0.0f);

    // B (4x16 f32) = all ones: invariant under any K/N striping convention.
    v2f b;
    b.x = 1.0f;
    b.y = 1.0f;

    // C (16x16 f32): C[0,n] = node5 sample 4, C[1,n] = node7 sample 4.
    v8f c = {};
    c[0] = (half == 0) ? n5_4 : 0.0f;  // row M=0, lanes 0-15
    c[1] = (half == 0) ? n7_4 : 0.0f;  // row M=1, lanes 0-15

    // D = A*B + C : D[0,n] = sum_k node5_k, D[1,n] = sum_k node7_k, exact in f32.
    // 8 args: (neg_a, A, neg_b, B, c_mod, C, reuse_a, reuse_b)
    v8f d = __builtin_amdgcn_wmma_f32_16x16x4_f32(
        false, a, false, b, (short)0, c, false, false);

    // Broadcast means in d[0]/d[1] of lanes 0-15 only. Multiply instead of
    // divide: avoids the ~15-op IEEE div expansion on the critical path
    // (<=1 ulp vs sum/5).
    float m5 = d[0] * 0.2f;
    float m7 = d[1] * 0.2f;

    // Root: out_i = LSE(lp0_i + m5, lp0_i + m7). Predication only after WMMA.
    if (l < 5) {
        float lp0 = lp[l * 3];
        out[l] = lse2(lp0 + m5, lp0 + m7);
    }
}

extern "C" void kernel_launch(void* const* d_in, const int* in_sizes, int n_in,
                              void* d_out, int out_size, void* d_ws, size_t ws_size,
                              hipStream_t stream) {
    (void)in_sizes; (void)n_in; (void)out_size; (void)d_ws; (void)ws_size;
    const float* lp = (const float*)d_in[0];  // [5,3] f32
    float* out = (float*)d_out;               // [1,5] f32
    circuit_wmma_kernel<<<1, 32, 0, stream>>>(lp, out);
}